// FilterDetectionsOwn_75093208203315
// MI455X (gfx1250) — compile-verified
//
#include <hip/hip_runtime.h>

// FilterDetections (per-class NMS + global top-K) for MI455X / gfx1250.
//
// Roofline: inputs are 67 MB total -> fully L2-resident (192 MB L2); the
// computation is a latency-bound chain of K=100 sequential NMS steps per
// (batch,class). CDNA5-specific paths used: async global->LDS staging of the
// per-(b,c) score column (global_load_async_to_lds_b32 + s_wait_asynccnt,
// ASYNCcnt-tracked), global_prefetch warming of the box array, wave32 shuffle
// reductions. WMMA does not apply (no matmul structure in IoU/top-k).
//
// 512 threads/block: per NMS step the critical path is N/THREADS serial
// LDS-load->IoU->compare elements; LDS capacity (100 KB score array) caps
// residency at 3 blocks/WGP either way, so wider blocks halve step latency
// at unchanged occupancy (3 blocks x 16 waves = 12 waves/SIMD).

#define B_ 8
#define N_ 25000
#define C_ 80
#define K_ 100
#define THREADS 512
#define NWAVES (THREADS / 32)
#define NEGF (-1e9f)
#define VALID_THR (-5e8f)   // reference: s[i] > NEG*0.5
#define SCORE_THR 0.05f

// Block-wide argmax with lowest-index tie-break (matches jnp.argmax /
// lax.top_k stability). wave32 shuffle reduce + cross-wave LDS combine.
__device__ __forceinline__ void block_argmax(float v, int idx,
                                             volatile float* red_s,
                                             volatile int* red_i,
                                             volatile float* bc_s,
                                             volatile int* bc_i) {
#pragma unroll
  for (int off = 16; off > 0; off >>= 1) {
    float ov = __shfl_xor(v, off, 32);
    int oi = __shfl_xor(idx, off, 32);
    if (ov > v || (ov == v && oi < idx)) { v = ov; idx = oi; }
  }
  const int lane = threadIdx.x & 31;
  const int wave = threadIdx.x >> 5;
  if (lane == 0) { red_s[wave] = v; red_i[wave] = idx; }
  __syncthreads();
  if (threadIdx.x == 0) {
    float bv = red_s[0];
    int bi = red_i[0];
#pragma unroll
    for (int w = 1; w < NWAVES; ++w) {
      float ws = red_s[w];
      int wi = red_i[w];
      if (ws > bv || (ws == bv && wi < bi)) { bv = ws; bi = wi; }
    }
    *bc_s = bv;
    *bc_i = bi;
  }
  __syncthreads();
}

// One block per (b,c) pair: sequential NMS (K steps) with scores resident in
// LDS (async-staged), boxes streamed from L2. Suppression is fused with the
// next iteration's argmax (single LDS pass per step).
__global__ void __launch_bounds__(THREADS)
nms_kernel(const float* __restrict__ boxes, const float* __restrict__ cls,
           float* __restrict__ cand_sc, int* __restrict__ cand_idx) {
  extern __shared__ float sc[];  // N_ floats (dynamic LDS, 100000 B)
  __shared__ float red_s[NWAVES];
  __shared__ int red_i[NWAVES];
  __shared__ float bc_s;
  __shared__ int bc_i;
  __shared__ float selbox[5];  // x1 y1 x2 y2 area

  const int b = blockIdx.x / C_;
  const int c = blockIdx.x - b * C_;
  const int tid = threadIdx.x;
  const float* clsbase = cls + (size_t)b * N_ * C_ + c;          // column base
  const float4* bx = reinterpret_cast<const float4*>(boxes) + (size_t)b * N_;

  // ---- CDNA5 async global->LDS staging of this (b,c) score column ----
  // Low 32 bits of a generic LDS address are the LDS byte offset (ISA 10.2).
  unsigned sc_base = (unsigned)(size_t)sc;
  for (int j = tid; j < N_; j += THREADS) {
    unsigned vo = (unsigned)j * (unsigned)(C_ * sizeof(float));  // GVS byte ofs
    unsigned la = sc_base + (unsigned)j * 4u;                    // LDS byte addr
    asm volatile("global_load_async_to_lds_b32 %0, %1, %2"
                 :: "v"(la), "v"(vo), "s"(clsbase)
                 : "memory");
  }
  // Warm L2 with this image's boxes while the async copies are in flight
  // (one prefetch per 128B line).
  for (int j = tid * 8; j < N_; j += THREADS * 8)
    __builtin_prefetch((const void*)(bx + j), 0, 1);
  asm volatile("s_wait_asynccnt 0x0" ::: "memory");
  __syncthreads();

  // ---- score-threshold prefilter + initial argmax ----
  float mv = NEGF;
  int mi = 0;
  for (int j = tid; j < N_; j += THREADS) {
    float s = sc[j];
    if (!(s > SCORE_THR)) { s = NEGF; sc[j] = NEGF; }
    if (s > mv) { mv = s; mi = j; }  // ascending j + strict '>' = first max
  }
  block_argmax(mv, mi, red_s, red_i, &bc_s, &bc_i);

  float* csc = cand_sc + (size_t)blockIdx.x * K_;
  int* cix = cand_idx + (size_t)blockIdx.x * K_;

  for (int k = 0; k < K_; ++k) {
    const float bs = bc_s;
    const int bi = bc_i;
    const bool valid = bs > VALID_THR;  // uniform across block
    if (tid == 0) { csc[k] = valid ? bs : NEGF; cix[k] = valid ? bi : 0; }
    if (!valid) {
      // All remaining steps select NEG -> fill and stop (matches reference:
      // invalid slots never contribute to outputs).
      if (tid == 0) {
        for (int kk = k + 1; kk < K_; ++kk) { csc[kk] = NEGF; cix[kk] = 0; }
      }
      break;  // uniform break
    }
    if (tid == 0) {
      float4 q = bx[bi];
      selbox[0] = q.x; selbox[1] = q.y; selbox[2] = q.z; selbox[3] = q.w;
      selbox[4] = (q.z - q.x) * (q.w - q.y);
      sc[bi] = NEGF;  // s2.at[i].set(NEG)
    }
    __syncthreads();
    const float x1 = selbox[0], y1 = selbox[1], x2 = selbox[2], y2 = selbox[3];
    const float ai = selbox[4];
    // Fused suppress + next-argmax pass.
    mv = NEGF;
    mi = 0;
    for (int j = tid; j < N_; j += THREADS) {
      float s = sc[j];
      if (s > VALID_THR) {  // skip already-suppressed (IoU write is a no-op)
        float4 q = bx[j];
        float xx1 = fmaxf(x1, q.x), yy1 = fmaxf(y1, q.y);
        float xx2 = fminf(x2, q.z), yy2 = fminf(y2, q.w);
        float inter = fmaxf(xx2 - xx1, 0.f) * fmaxf(yy2 - yy1, 0.f);
        float aj = (q.z - q.x) * (q.w - q.y);
        float iou = inter / fmaxf(ai + aj - inter, 1e-8f);
        if (iou > 0.5f) { s = NEGF; sc[j] = NEGF; }
      }
      if (s > mv) { mv = s; mi = j; }
    }
    block_argmax(mv, mi, red_s, red_i, &bc_s, &bc_i);
  }
}

// One block per image: stable top-K over the C_*K_ = 8000 candidates
// (LDS-resident), emit boxes/scores/labels/n_valid.
// d_out layout (flat float, reference return order):
//   [0, B*K*4)           boxes
//   [B*K*4, +B*K)        scores
//   [.., +B*K)           labels (int cast to float)
//   [.., +B)             n_valid (int cast to float)
__global__ void __launch_bounds__(THREADS)
topk_kernel(const float* __restrict__ boxes, const float* __restrict__ cand_sc,
            const int* __restrict__ cand_idx, float* __restrict__ out) {
  __shared__ float cs[C_ * K_];  // 32 KB
  __shared__ float red_s[NWAVES];
  __shared__ int red_i[NWAVES];
  __shared__ float bc_s;
  __shared__ int bc_i;

  const int b = blockIdx.x;
  const int tid = threadIdx.x;
  const float* csg = cand_sc + (size_t)b * C_ * K_;
  for (int f = tid; f < C_ * K_; f += THREADS) cs[f] = csg[f];
  __syncthreads();

  float* ob = out;
  float* os = out + (size_t)B_ * K_ * 4;
  float* ol = os + (size_t)B_ * K_;
  float* on = ol + (size_t)B_ * K_;
  const float4* bx = reinterpret_cast<const float4*>(boxes) + (size_t)b * N_;
  int nv = 0;

  for (int k = 0; k < K_; ++k) {
    float mv = -2e9f;  // below NEGF so exhausted entries never win
    int mi = 0;
    for (int f = tid; f < C_ * K_; f += THREADS) {
      float s = cs[f];
      if (s > mv) { mv = s; mi = f; }
    }
    block_argmax(mv, mi, red_s, red_i, &bc_s, &bc_i);
    if (tid == 0) {
      const float s = bc_s;
      const int f = bc_i;
      const bool v = s > VALID_THR;
      const size_t ok = (size_t)b * K_ + k;
      if (v) {
        const int idx = cand_idx[(size_t)b * C_ * K_ + f];
        const float4 q = bx[idx];
        ob[ok * 4 + 0] = q.x; ob[ok * 4 + 1] = q.y;
        ob[ok * 4 + 2] = q.z; ob[ok * 4 + 3] = q.w;
        os[ok] = s;
        ol[ok] = (float)(f / K_);  // label = class id
        ++nv;
      } else {
        ob[ok * 4 + 0] = -1.f; ob[ok * 4 + 1] = -1.f;
        ob[ok * 4 + 2] = -1.f; ob[ok * 4 + 3] = -1.f;
        os[ok] = -1.f;
        ol[ok] = -1.f;
      }
      cs[f] = -2e9f;  // remove selected candidate
    }
    __syncthreads();
  }
  if (tid == 0) on[b] = (float)nv;
}

extern "C" void kernel_launch(void* const* d_in, const int* in_sizes, int n_in,
                              void* d_out, int out_size, void* d_ws,
                              size_t ws_size, hipStream_t stream) {
  (void)in_sizes; (void)n_in; (void)out_size; (void)ws_size;
  const float* boxes = (const float*)d_in[0];  // (B,N,4) f32
  const float* cls = (const float*)d_in[1];    // (B,N,C) f32

  // Workspace: per-(b,c,k) candidate score + box index (512 KB total).
  float* cand_sc = (float*)d_ws;
  int* cand_idx = (int*)((char*)d_ws + (size_t)B_ * C_ * K_ * sizeof(float));

  const size_t dyn = (size_t)N_ * sizeof(float);  // 100000 B dynamic LDS
  (void)hipFuncSetAttribute((const void*)nms_kernel,
                            hipFuncAttributeMaxDynamicSharedMemorySize,
                            (int)dyn);

  nms_kernel<<<B_ * C_, THREADS, dyn, stream>>>(boxes, cls, cand_sc, cand_idx);
  topk_kernel<<<B_, THREADS, 0, stream>>>(boxes, cand_sc, cand_idx,
                                          (float*)d_out);
}